// LightGCN_86431921865202
// MI455X (gfx1250) — compile-verified
//
#include <hip/hip_runtime.h>
#include <math.h>

typedef __attribute__((ext_vector_type(2))) float v2f;
typedef __attribute__((ext_vector_type(8))) float v8f;

#define EMB_D   128
#define EMB_DV4 32   // EMB_D / 4

// ---------------------------------------------------------------------------
// 1) cur = concat(user_emb, item_emb); also zero the reg-loss scalar.
// ---------------------------------------------------------------------------
__global__ void lg_init_emb(const float4* __restrict__ uemb,
                            const float4* __restrict__ iemb,
                            float4* __restrict__ cur,
                            float* __restrict__ regp,
                            int nUser4, int nTot4) {
    int i = blockIdx.x * blockDim.x + threadIdx.x;
    if (i == 0) regp[0] = 0.0f;
    if (i >= nTot4) return;
    cur[i] = (i < nUser4) ? uemb[i] : iemb[i - nUser4];
}

// ---------------------------------------------------------------------------
// 2) Batch accumulators = ego (layer-0) embeddings; wave-reduced L1 sums
//    atomically folded into the single reg scalar.
// ---------------------------------------------------------------------------
__global__ void lg_init_batch(const float4* __restrict__ uemb,
                              const float4* __restrict__ iemb,
                              const int* __restrict__ user,
                              const int* __restrict__ pos,
                              const int* __restrict__ neg,
                              float4* __restrict__ uacc,
                              float4* __restrict__ pacc,
                              float4* __restrict__ nacc,
                              float* __restrict__ regp, int B) {
    int t = blockIdx.x * blockDim.x + threadIdx.x;
    int b = t >> 5, c = t & 31;
    if (b >= B) return;
    float4 a = uemb[(long)user[b] * EMB_DV4 + c];
    float4 p = iemb[(long)pos[b]  * EMB_DV4 + c];
    float4 n = iemb[(long)neg[b]  * EMB_DV4 + c];
    uacc[(long)b * EMB_DV4 + c] = a;
    pacc[(long)b * EMB_DV4 + c] = p;
    nacc[(long)b * EMB_DV4 + c] = n;
    float s = fabsf(a.x) + fabsf(a.y) + fabsf(a.z) + fabsf(a.w)
            + fabsf(p.x) + fabsf(p.y) + fabsf(p.z) + fabsf(p.w)
            + fabsf(n.x) + fabsf(n.y) + fabsf(n.z) + fabsf(n.w);
    for (int off = 16; off > 0; off >>= 1) s += __shfl_down(s, off, 32);
    if ((threadIdx.x & 31) == 0) atomicAdd(regp, s);
}

// ---------------------------------------------------------------------------
// 3) Zero the scatter destination.
// ---------------------------------------------------------------------------
__global__ void lg_zero(float4* __restrict__ p, int n4) {
    int i = blockIdx.x * blockDim.x + threadIdx.x;
    if (i < n4) p[i] = make_float4(0.f, 0.f, 0.f, 0.f);
}

// ---------------------------------------------------------------------------
// 4) SpMM scatter: one wave per edge, lane c owns dwords [4c,4c+4).
//    Coalesced 512B row gather; 4 global_atomic_add_f32 per lane.
//    Table (76.8MB) + dest (76.8MB) fit in the 192MB L2 -> atomics stay in L2.
// ---------------------------------------------------------------------------
__global__ void lg_spmm(const float* __restrict__ cur,
                        float* __restrict__ nxt,
                        const int* __restrict__ erow,
                        const int* __restrict__ ecol,
                        const float* __restrict__ eval_, int E) {
    long t = (long)blockIdx.x * blockDim.x + threadIdx.x;
    int e = (int)(t >> 5);
    int c = (int)(t & 31);
    if (e >= E) return;
    int   col = ecol[e];
    int   row = erow[e];
    float v   = eval_[e];
    float4 m = ((const float4*)(cur + (long)col * EMB_D))[c];
    float* dst = nxt + (long)row * EMB_D + c * 4;
    atomicAdd(dst + 0, m.x * v);
    atomicAdd(dst + 1, m.y * v);
    atomicAdd(dst + 2, m.z * v);
    atomicAdd(dst + 3, m.w * v);
}

// ---------------------------------------------------------------------------
// 5) Fold this layer's output rows into the batch accumulators.
// ---------------------------------------------------------------------------
__global__ void lg_gather(const float4* __restrict__ nxt,
                          const int* __restrict__ user,
                          const int* __restrict__ pos,
                          const int* __restrict__ neg,
                          float4* __restrict__ uacc,
                          float4* __restrict__ pacc,
                          float4* __restrict__ nacc,
                          int B, int nUsers) {
    int t = blockIdx.x * blockDim.x + threadIdx.x;
    int b = t >> 5, c = t & 31;
    if (b >= B) return;
    long o = (long)b * EMB_DV4 + c;
    float4 a = nxt[(long)user[b] * EMB_DV4 + c];
    float4 p = nxt[(long)(nUsers + pos[b]) * EMB_DV4 + c];
    float4 n = nxt[(long)(nUsers + neg[b]) * EMB_DV4 + c];
    float4 A = uacc[o]; A.x += a.x; A.y += a.y; A.z += a.z; A.w += a.w; uacc[o] = A;
    float4 P = pacc[o]; P.x += p.x; P.y += p.y; P.z += p.z; P.w += p.w; pacc[o] = P;
    float4 Nn = nacc[o]; Nn.x += n.x; Nn.y += n.y; Nn.z += n.z; Nn.w += n.w; nacc[o] = Nn;
}

// ---------------------------------------------------------------------------
// 6) Scoring via V_WMMA_F32_16X16X4_F32: per wave, 16 batch rows.
//    D = A(16x4) * B(4x16) + C computes all pairwise dots; diagonal = scores.
//    A layout: lane<16 -> K=k0+{0,1}; lane>=16 -> K=k0+{2,3}, row M = lane&15.
//    B layout is the identical load pattern with N = lane&15.
//    Diagonal: D[r][r] at (vgpr r, lane r); D[8+r][8+r] at (vgpr r, lane 24+r).
// ---------------------------------------------------------------------------
__global__ void lg_score(const float* __restrict__ uacc,
                         const float* __restrict__ pacc,
                         const float* __restrict__ nacc,
                         const float* __restrict__ regp,
                         float* __restrict__ out, int B) {
    int lane = threadIdx.x & 31;
    int wave = (blockIdx.x * blockDim.x + threadIdx.x) >> 5;
    int base = wave * 16;
    if (base >= B) return;

    int row   = base + (lane & 15);
    int khalf = (lane >> 4) * 2;                 // 0 or 2
    const float* urow = uacc + (long)row * EMB_D;
    const float* prow = pacc + (long)row * EMB_D;
    const float* nrow = nacc + (long)row * EMB_D;

    v8f cp = {};
    v8f cn = {};
    #pragma unroll 4
    for (int k0 = 0; k0 < EMB_D; k0 += 4) {
        int k = k0 + khalf;                      // even -> 8B aligned
        v2f a  = *(const v2f*)(urow + k);
        v2f bp = *(const v2f*)(prow + k);
        v2f bn = *(const v2f*)(nrow + k);
        cp = __builtin_amdgcn_wmma_f32_16x16x4_f32(false, a, false, bp,
                                                   (short)0, cp, false, false);
        cn = __builtin_amdgcn_wmma_f32_16x16x4_f32(false, a, false, bn,
                                                   (short)0, cn, false, false);
    }

    // Extract the diagonal the wave owns.
    float ps = 0.f, ns = 0.f;
    bool have = false;
    int  oidx = 0;
    #pragma unroll
    for (int r = 0; r < 8; ++r) {
        if (lane == r)      { ps = cp[r]; ns = cn[r]; have = true; oidx = base + r; }
        if (lane == 24 + r) { ps = cp[r]; ns = cn[r]; have = true; oidx = base + 8 + r; }
    }
    if (have) {
        // latents are acc/4 -> score = raw_dot / 16
        float x  = (ns - ps) * (1.0f / 16.0f);
        float sp = (x > 0.f) ? (x + log1pf(expf(-x))) : log1pf(expf(x));
        out[oidx] = sp + 1e-4f * regp[0];
    }
}

// ---------------------------------------------------------------------------
extern "C" void kernel_launch(void* const* d_in, const int* in_sizes, int n_in,
                              void* d_out, int out_size, void* d_ws, size_t ws_size,
                              hipStream_t stream) {
    const float* uemb  = (const float*)d_in[0];
    const float* iemb  = (const float*)d_in[1];
    const int*   erow  = (const int*)  d_in[2];
    const int*   ecol  = (const int*)  d_in[3];
    const float* eval_ = (const float*)d_in[4];
    const int*   user  = (const int*)  d_in[5];
    const int*   pos   = (const int*)  d_in[6];
    const int*   neg   = (const int*)  d_in[7];
    float*       out   = (float*)d_out;

    const int nUsers = in_sizes[0] / EMB_D;
    const int nItems = in_sizes[1] / EMB_D;
    const int N      = nUsers + nItems;
    const int E      = in_sizes[2];
    const int B      = in_sizes[5];

    // Workspace layout (fp32): two ping-pong table buffers + batch accs + scalar.
    float* ws   = (float*)d_ws;
    float* cur  = ws;
    float* nxt  = cur  + (size_t)N * EMB_D;
    float* uacc = nxt  + (size_t)N * EMB_D;
    float* pacc = uacc + (size_t)B * EMB_D;
    float* nacc = pacc + (size_t)B * EMB_D;
    float* regp = nacc + (size_t)B * EMB_D;

    const int BS = 256;
    const int nUser4 = nUsers * EMB_DV4;
    const int nTot4  = N * EMB_DV4;

    lg_init_emb<<<(nTot4 + BS - 1) / BS, BS, 0, stream>>>(
        (const float4*)uemb, (const float4*)iemb, (float4*)cur, regp, nUser4, nTot4);

    lg_init_batch<<<(B * 32 + BS - 1) / BS, BS, 0, stream>>>(
        (const float4*)uemb, (const float4*)iemb, user, pos, neg,
        (float4*)uacc, (float4*)pacc, (float4*)nacc, regp, B);

    for (int layer = 0; layer < 3; ++layer) {
        lg_zero<<<(nTot4 + BS - 1) / BS, BS, 0, stream>>>((float4*)nxt, nTot4);

        long tE = (long)E * 32;
        lg_spmm<<<(int)((tE + BS - 1) / BS), BS, 0, stream>>>(
            cur, nxt, erow, ecol, eval_, E);

        lg_gather<<<(B * 32 + BS - 1) / BS, BS, 0, stream>>>(
            (const float4*)nxt, user, pos, neg,
            (float4*)uacc, (float4*)pacc, (float4*)nacc, B, nUsers);

        float* tmp = cur; cur = nxt; nxt = tmp;   // deterministic host-side swap
    }

    int waves = (B + 15) / 16;                    // 512 waves, 16 rows each
    lg_score<<<(waves * 32 + BS - 1) / BS, BS, 0, stream>>>(
        uacc, pacc, nacc, regp, out, B);
}